// DualMambaBlock_68247030333815
// MI455X (gfx1250) — compile-verified
//
#include <hip/hip_runtime.h>
#include <math.h>

#define B_ 16
#define K_ 1024
#define D_ 512
#define S_ 16
#define ROWS_ (B_ * K_)   // 16384 tokens

typedef float v2f __attribute__((ext_vector_type(2)));
typedef float v8f __attribute__((ext_vector_type(8)));

// ---------------------------------------------------------------------------
// CDNA5 WMMA f32 16x16x4:  D(16x16,f32) = A(16x4,f32) x B(4x16,f32) + C
// A frag: lane l<16 -> row M=l, v={K0,K1}; l>=16 -> row M=l-16, v={K2,K3}
// B frag: lane l<16 -> col N=l, v={K0,K1}; l>=16 -> col N=l-16, v={K2,K3}
// C/D:    vgpr g: lanes 0-15 (M=g,N=lane), lanes 16-31 (M=8+g,N=lane-16)
// ---------------------------------------------------------------------------
__device__ __forceinline__ v8f wmma_f32(v2f a, v2f b, v8f c) {
  return __builtin_amdgcn_wmma_f32_16x16x4_f32(
      /*neg_a=*/false, a, /*neg_b=*/false, b,
      /*c_mod=*/(short)0, c, /*reuse_a=*/false, /*reuse_b=*/false);
}

__device__ __forceinline__ float silu_f(float x) {
  return x / (1.f + __expf(-x));
}

// LDS byte offset of a __shared__ object (low 32 bits of the shared-aperture
// generic address == workgroup-relative LDS address).
__device__ __forceinline__ unsigned lds_off(const void* p) {
  return (unsigned)(size_t)p;
}

// Async copy of 16 contiguous bytes global -> LDS (tracked by ASYNCcnt).
__device__ __forceinline__ void async_copy_b128(unsigned lds_dst, const void* gsrc) {
  asm volatile("global_load_async_to_lds_b128 %0, %1, off"
               :: "v"(lds_dst), "v"((unsigned long long)(size_t)gsrc)
               : "memory");
}
__device__ __forceinline__ void async_wait0() {
  asm volatile("s_wait_asynccnt 0x0" ::: "memory");
}

// ---------------------------------------------------------------------------
// Generic WMMA GEMM:  Y[M,N] = act(X[M,Kd] @ W[Kd,N] + bias)
// W column panel (Kd x 16) async-staged into LDS, shared by all 8 waves.
// Each wave computes a 64x16 tile (4 accumulators sharing one B fragment).
// Register double-buffering: k+4 fragments are fetched before the k wmma
// burst so s_wait_loadcnt overlaps the matrix pipe.
// grid = (N/16, M/512), block = 256
// ---------------------------------------------------------------------------
__global__ __launch_bounds__(256) void gemm_wmma_k(
    const float* __restrict__ X, int ldx,
    const float* __restrict__ W, int ldw,
    const float* __restrict__ bias,
    float* __restrict__ Y, int ldy, int Kd, int act) {
  __shared__ float wpanel[D_ * 16];            // up to 32 KB
  const int tid = threadIdx.x;
  const int lane = tid & 31;
  const int wave = tid >> 5;
  const int n0 = blockIdx.x * 16;

  // ---- async stage W[0:Kd, n0:n0+16] into LDS (row-major k x 16) ----
  const int nvec = Kd * 4;                     // float4 count
  for (int v = tid; v < nvec; v += 256) {
    const int kk = v >> 2;                     // panel row
    const int nc = (v & 3) << 2;               // col within row
    async_copy_b128(lds_off(&wpanel[v << 2]),
                    W + (size_t)kk * ldw + n0 + nc);
  }
  async_wait0();
  __syncthreads();

  const int l15 = lane & 15;
  const int koff = (lane >> 4) << 1;           // 0 or 2
  const int m0 = (blockIdx.y * 8 + wave) * 64;
  const float* x0 = X + (size_t)(m0 + 0  + l15) * ldx + koff;
  const float* x1 = X + (size_t)(m0 + 16 + l15) * ldx + koff;
  const float* x2 = X + (size_t)(m0 + 32 + l15) * ldx + koff;
  const float* x3 = X + (size_t)(m0 + 48 + l15) * ldx + koff;
  v8f c0 = {}, c1 = {}, c2 = {}, c3 = {};

  v2f b, a0v, a1v, a2v, a3v;
  b.x = wpanel[koff * 16 + l15];
  b.y = wpanel[(koff + 1) * 16 + l15];
  a0v = *(const v2f*)(x0);
  a1v = *(const v2f*)(x1);
  a2v = *(const v2f*)(x2);
  a3v = *(const v2f*)(x3);
  for (int kk = 0; kk + 4 < Kd; kk += 4) {
    const int kn = kk + 4;
    v2f nb, n0v, n1v, n2v, n3v;
    nb.x = wpanel[(kn + koff) * 16 + l15];
    nb.y = wpanel[(kn + koff + 1) * 16 + l15];
    n0v = *(const v2f*)(x0 + kn);
    n1v = *(const v2f*)(x1 + kn);
    n2v = *(const v2f*)(x2 + kn);
    n3v = *(const v2f*)(x3 + kn);
    c0 = wmma_f32(a0v, b, c0);
    c1 = wmma_f32(a1v, b, c1);
    c2 = wmma_f32(a2v, b, c2);
    c3 = wmma_f32(a3v, b, c3);
    b = nb; a0v = n0v; a1v = n1v; a2v = n2v; a3v = n3v;
  }
  c0 = wmma_f32(a0v, b, c0);
  c1 = wmma_f32(a1v, b, c1);
  c2 = wmma_f32(a2v, b, c2);
  c3 = wmma_f32(a3v, b, c3);

  const float bn = bias ? bias[n0 + l15] : 0.f;
  const int col = n0 + l15;
  const int rbase = m0 + (lane >> 4) * 8;
  auto store_tile = [&](v8f cc, int moff) {
#pragma unroll
    for (int g = 0; g < 8; ++g) {
      float v = cc[g] + bn;
      if (act == 1) v = silu_f(v);
      Y[(size_t)(rbase + moff + g) * ldy + col] = v;
    }
  };
  store_tile(c0, 0);
  store_tile(c1, 16);
  store_tile(c2, 32);
  store_tile(c3, 48);
}

// ---------------------------------------------------------------------------
// Batched Gram matrix: G[b] = X[b] @ X[b]^T  (X: B x K_ x C row-major)
// A row panel (16 x C) async-staged into LDS, shared by all 8 waves.
// Each wave computes a 16x64 tile (4 accumulators sharing one A fragment),
// with register double-buffering of the global B fragments.
// grid = (K_/512, K_/16, B_), block = 256.  lc = log2(C).
// ---------------------------------------------------------------------------
__global__ __launch_bounds__(256) void gram_wmma_k(
    const float* __restrict__ X, int C, int lc, float* __restrict__ G) {
  __shared__ float apanel[16 * D_];            // up to 32 KB
  const int b = blockIdx.z;
  const float* Xb = X + (size_t)b * K_ * C;
  float* Gb = G + (size_t)b * K_ * K_;
  const int tid = threadIdx.x;
  const int lane = tid & 31;
  const int wave = tid >> 5;
  const int i0 = blockIdx.y * 16;

  // ---- async stage X[i0:i0+16, :] into LDS (row-major 16 x C) ----
  const int nvec = (16 * C) >> 2;
  for (int v = tid; v < nvec; v += 256) {
    const int fl = v << 2;
    const int r = fl >> lc;                    // row 0..15
    const int c = fl & (C - 1);
    async_copy_b128(lds_off(&apanel[fl]),
                    Xb + (size_t)(i0 + r) * C + c);
  }
  async_wait0();
  __syncthreads();

  const int l15 = lane & 15;
  const int koff = (lane >> 4) << 1;
  const int j0 = (blockIdx.x * 8 + wave) * 64;
  const float* b0 = Xb + (size_t)(j0 + 0  + l15) * C + koff;
  const float* b1 = Xb + (size_t)(j0 + 16 + l15) * C + koff;
  const float* b2 = Xb + (size_t)(j0 + 32 + l15) * C + koff;
  const float* b3 = Xb + (size_t)(j0 + 48 + l15) * C + koff;
  const float* arow = &apanel[l15 * C + koff];
  v8f c0 = {}, c1 = {}, c2 = {}, c3 = {};

  v2f a = *(const v2f*)(arow);
  v2f f0 = *(const v2f*)(b0);
  v2f f1 = *(const v2f*)(b1);
  v2f f2 = *(const v2f*)(b2);
  v2f f3 = *(const v2f*)(b3);
  for (int kk = 0; kk + 4 < C; kk += 4) {
    const int kn = kk + 4;
    v2f na = *(const v2f*)(arow + kn);         // ds_load_b64
    v2f n0v = *(const v2f*)(b0 + kn);
    v2f n1v = *(const v2f*)(b1 + kn);
    v2f n2v = *(const v2f*)(b2 + kn);
    v2f n3v = *(const v2f*)(b3 + kn);
    c0 = wmma_f32(a, f0, c0);
    c1 = wmma_f32(a, f1, c1);
    c2 = wmma_f32(a, f2, c2);
    c3 = wmma_f32(a, f3, c3);
    a = na; f0 = n0v; f1 = n1v; f2 = n2v; f3 = n3v;
  }
  c0 = wmma_f32(a, f0, c0);
  c1 = wmma_f32(a, f1, c1);
  c2 = wmma_f32(a, f2, c2);
  c3 = wmma_f32(a, f3, c3);

  const int irow = i0 + (lane >> 4) * 8;
  auto store_tile = [&](v8f cc, int joff) {
    const int col = j0 + joff + l15;
#pragma unroll
    for (int g = 0; g < 8; ++g)
      Gb[(size_t)(irow + g) * K_ + col] = cc[g];
  };
  store_tile(c0, 0);
  store_tile(c1, 16);
  store_tile(c2, 32);
  store_tile(c3, 48);
}

// ---------------------------------------------------------------------------
// Row squared norms (one wave per row). grid = rows/8, block = 256
// ---------------------------------------------------------------------------
__global__ __launch_bounds__(256) void row_norms_k(
    const float* __restrict__ X, int C, float* __restrict__ n) {
  const int row = blockIdx.x * 8 + (threadIdx.x >> 5);
  const int lane = threadIdx.x & 31;
  const float* xr = X + (size_t)row * C;
  float s = 0.f;
  for (int c = lane; c < C; c += 32) { float v = xr[c]; s += v * v; }
#pragma unroll
  for (int off = 16; off; off >>= 1) s += __shfl_xor(s, off, 32);
  if (lane == 0) n[row] = s;
}

// ---------------------------------------------------------------------------
// Farthest-point ordering from Gram matrix. One block of 1024 per batch.
// dist(i,t) = n_i + n_t - 2*G[t][i]  (G symmetric -> contiguous row reads)
// ---------------------------------------------------------------------------
__global__ __launch_bounds__(1024) void fps_order_k(
    const float* __restrict__ G, const float* __restrict__ nrm,
    int* __restrict__ ord) {
  const int b = blockIdx.x;
  const float* Gb = G + (size_t)b * K_ * K_;
  const float* nb = nrm + (size_t)b * K_;
  int* ob = ord + (size_t)b * K_;
  const int i = threadIdx.x;
  const int lane = i & 31, wave = i >> 5;
  __shared__ float sd[32];
  __shared__ int si[32];
  __shared__ int sfar;
  const float ni = nb[i];
  float dist = ni + nb[0] - 2.f * Gb[i];
  if (i == 0) ob[0] = 0;
  for (int s = 1; s < K_; ++s) {
    float d = dist; int idx = i;                       // argmax, first-index tie
#pragma unroll
    for (int off = 16; off; off >>= 1) {
      float od = __shfl_xor(d, off, 32);
      int oi = __shfl_xor(idx, off, 32);
      if (od > d || (od == d && oi < idx)) { d = od; idx = oi; }
    }
    if (lane == 0) { sd[wave] = d; si[wave] = idx; }
    __syncthreads();
    if (wave == 0) {
      d = sd[lane]; idx = si[lane];
#pragma unroll
      for (int off = 16; off; off >>= 1) {
        float od = __shfl_xor(d, off, 32);
        int oi = __shfl_xor(idx, off, 32);
        if (od > d || (od == d && oi < idx)) { d = od; idx = oi; }
      }
      if (lane == 0) { sfar = idx; ob[s] = idx; }
    }
    __syncthreads();
    const int far = sfar;
    const float nd = ni + nb[far] - 2.f * Gb[(size_t)far * K_ + i];
    dist = fminf(dist, nd);
  }
}

// ---------------------------------------------------------------------------
// Nearest-point ordering from Gram matrix. One block of 1024 per batch.
// ---------------------------------------------------------------------------
__global__ __launch_bounds__(1024) void nps_order_k(
    const float* __restrict__ G, const float* __restrict__ nrm,
    int* __restrict__ ord) {
  const int b = blockIdx.x;
  const float* Gb = G + (size_t)b * K_ * K_;
  const float* nb = nrm + (size_t)b * K_;
  int* ob = ord + (size_t)b * K_;
  const int i = threadIdx.x;
  const int lane = i & 31, wave = i >> 5;
  __shared__ float sd[32];
  __shared__ int si[32];
  __shared__ int snxt;
  const float ni = nb[i];
  bool used = (i == 0);
  int cur = 0;
  if (i == 0) ob[0] = 0;
  for (int s = 1; s < K_; ++s) {
    float d = used ? __builtin_inff()
                   : (ni + nb[cur] - 2.f * Gb[(size_t)cur * K_ + i]);
    int idx = i;                                       // argmin, first-index tie
#pragma unroll
    for (int off = 16; off; off >>= 1) {
      float od = __shfl_xor(d, off, 32);
      int oi = __shfl_xor(idx, off, 32);
      if (od < d || (od == d && oi < idx)) { d = od; idx = oi; }
    }
    if (lane == 0) { sd[wave] = d; si[wave] = idx; }
    __syncthreads();
    if (wave == 0) {
      d = sd[lane]; idx = si[lane];
#pragma unroll
      for (int off = 16; off; off >>= 1) {
        float od = __shfl_xor(d, off, 32);
        int oi = __shfl_xor(idx, off, 32);
        if (od < d || (od == d && oi < idx)) { d = od; idx = oi; }
      }
      if (lane == 0) { snxt = idx; ob[s] = idx; }
    }
    __syncthreads();
    cur = snxt;
    if (i == cur) used = true;
  }
}

// ---------------------------------------------------------------------------
// Gather rows by order: Y[b][k][:] = X[b][ord[b][k]][:].  grid = B_*K_
// ---------------------------------------------------------------------------
__global__ void gather_rows_k(const float* __restrict__ X,
                              const int* __restrict__ ord,
                              float* __restrict__ Y, int C) {
  const int r = blockIdx.x;          // 0..ROWS_-1
  const int b = r >> 10;             // / K_
  const int src = ord[r];
  const float* xs = X + ((size_t)b * K_ + src) * C;
  float* yd = Y + (size_t)r * C;
  for (int c = threadIdx.x; c < C; c += blockDim.x) yd[c] = xs[c];
}

// ---------------------------------------------------------------------------
// LayerNorm over C=512, one wave per token. grid = ROWS_/8, block = 256
// ---------------------------------------------------------------------------
__global__ __launch_bounds__(256) void layernorm_k(
    const float* __restrict__ X, const float* __restrict__ g,
    const float* __restrict__ be, float* __restrict__ Y) {
  const int row = blockIdx.x * 8 + (threadIdx.x >> 5);
  const int lane = threadIdx.x & 31;
  const float* xr = X + (size_t)row * D_;
  float vals[16];
  float s = 0.f;
#pragma unroll
  for (int t = 0; t < 16; ++t) { vals[t] = xr[lane + 32 * t]; s += vals[t]; }
#pragma unroll
  for (int off = 16; off; off >>= 1) s += __shfl_xor(s, off, 32);
  const float mu = s * (1.f / (float)D_);
  float v = 0.f;
#pragma unroll
  for (int t = 0; t < 16; ++t) { float dd = vals[t] - mu; v += dd * dd; }
#pragma unroll
  for (int off = 16; off; off >>= 1) v += __shfl_xor(v, off, 32);
  const float inv = rsqrtf(v * (1.f / (float)D_) + 1e-5f);
  float* yr = Y + (size_t)row * D_;
#pragma unroll
  for (int t = 0; t < 16; ++t) {
    const int c = lane + 32 * t;
    yr[c] = (vals[t] - mu) * inv * g[c] + be[c];
  }
}

// ---------------------------------------------------------------------------
// Fused head GEMM: S[r][0:16]=zc@Wb+bb, [16:32]=zc@Wc+bc,
//                  [32:48]=zc@Wd+bd,    [48:64]=zc@Wx+bx
// grid = ROWS_/128, block 256 (8 waves, 16 tokens each, 4 wmma accumulators)
// Register double-buffering of A and the four B fragments.
// ---------------------------------------------------------------------------
__global__ __launch_bounds__(256) void head_gemm_k(
    const float* __restrict__ ZC,
    const float* __restrict__ Wb, const float* __restrict__ bb,
    const float* __restrict__ Wc, const float* __restrict__ bc,
    const float* __restrict__ Wd, const float* __restrict__ bd,
    const float* __restrict__ Wx, const float* __restrict__ bx,
    float* __restrict__ S) {
  const int lane = threadIdx.x & 31;
  const int wave = threadIdx.x >> 5;
  const int m0 = (blockIdx.x * 8 + wave) * 16;
  const int l15 = lane & 15;
  const int koff = (lane >> 4) << 1;
  const float* xrow = ZC + (size_t)(m0 + l15) * D_ + koff;
  v8f a0 = {}, a1 = {}, a2 = {}, a3 = {};

  v2f a = *(const v2f*)(xrow);
  int r0 = koff * S_ + l15;
  v2f vb, vc, vd, vx;
  vb.x = Wb[r0]; vb.y = Wb[r0 + S_];
  vc.x = Wc[r0]; vc.y = Wc[r0 + S_];
  vd.x = Wd[r0]; vd.y = Wd[r0 + S_];
  vx.x = Wx[r0]; vx.y = Wx[r0 + S_];
  for (int kk = 0; kk + 4 < D_; kk += 4) {
    const int rn = (kk + 4 + koff) * S_ + l15;
    v2f na = *(const v2f*)(xrow + kk + 4);
    v2f nb_, nc_, nd_, nx_;
    nb_.x = Wb[rn]; nb_.y = Wb[rn + S_];
    nc_.x = Wc[rn]; nc_.y = Wc[rn + S_];
    nd_.x = Wd[rn]; nd_.y = Wd[rn + S_];
    nx_.x = Wx[rn]; nx_.y = Wx[rn + S_];
    a0 = wmma_f32(a, vb, a0);
    a1 = wmma_f32(a, vc, a1);
    a2 = wmma_f32(a, vd, a2);
    a3 = wmma_f32(a, vx, a3);
    a = na; vb = nb_; vc = nc_; vd = nd_; vx = nx_;
  }
  a0 = wmma_f32(a, vb, a0);
  a1 = wmma_f32(a, vc, a1);
  a2 = wmma_f32(a, vd, a2);
  a3 = wmma_f32(a, vx, a3);

  const int mrow = m0 + (lane >> 4) * 8;
  const int col = l15;
#pragma unroll
  for (int g = 0; g < 8; ++g) {
    const size_t r = (size_t)(mrow + g) * 64;
    S[r + col]      = a0[g] + bb[col];
    S[r + 16 + col] = a1[g] + bc[col];
    S[r + 32 + col] = a2[g] + bd[col];
    S[r + 48 + col] = a3[g] + bx[col];
  }
}

// ---------------------------------------------------------------------------
// Head pointwise: softmax(delta), y = Cs*(delta*Bs*xp), out = silu(y)*zc[:,c]
// (channels >= 16 of the branch output are silu(0)*zc = 0, so only 16 kept)
// ---------------------------------------------------------------------------
__global__ void head_pointwise_k(const float* __restrict__ S,
                                 const float* __restrict__ ZC,
                                 float* __restrict__ Y16) {
  const int id = blockIdx.x * blockDim.x + threadIdx.x;   // ROWS_*16 total
  if (id >= ROWS_ * S_) return;
  const int r = id >> 4, c = id & 15;
  const float* sp = S + (size_t)r * 64;
  const float Bs = sp[c], Cs = sp[16 + c], dl = sp[32 + c], xp = sp[48 + c];
  float mx = -__builtin_inff();
#pragma unroll
  for (int j = 0; j < 16; ++j) mx = fmaxf(mx, sp[32 + j]);
  float den = 0.f;
#pragma unroll
  for (int j = 0; j < 16; ++j) den += __expf(sp[32 + j] - mx);
  const float delta = __expf(dl - mx) / den;
  const float y = Cs * (delta * Bs * xp);
  Y16[(size_t)r * S_ + c] = silu_f(y) * ZC[(size_t)r * D_ + c];
}

// yF = 0.5*(yF + yN)
__global__ void mix_k(float* __restrict__ yF, const float* __restrict__ yN) {
  const int id = blockIdx.x * blockDim.x + threadIdx.x;
  if (id < ROWS_ * S_) yF[id] = 0.5f * (yF[id] + yN[id]);
}

// ---------------------------------------------------------------------------
extern "C" void kernel_launch(void* const* d_in, const int* in_sizes, int n_in,
                              void* d_out, int out_size, void* d_ws, size_t ws_size,
                              hipStream_t stream) {
  const float* tokens  = (const float*)d_in[0];
  const float* ln_f_g  = (const float*)d_in[1];
  const float* ln_f_b  = (const float*)d_in[2];
  const float* ln_n_g  = (const float*)d_in[3];
  const float* ln_n_b  = (const float*)d_in[4];
  const float* W_lin_f = (const float*)d_in[5];
  const float* b_lin_f = (const float*)d_in[6];
  const float* W_lin_n = (const float*)d_in[7];
  const float* b_lin_n = (const float*)d_in[8];
  const float* W_conv_f= (const float*)d_in[9];
  const float* b_conv_f= (const float*)d_in[10];
  const float* W_conv_n= (const float*)d_in[11];
  const float* b_conv_n= (const float*)d_in[12];
  const float* W_b_f   = (const float*)d_in[13];
  const float* b_b_f   = (const float*)d_in[14];
  const float* W_c_f   = (const float*)d_in[15];
  const float* b_c_f   = (const float*)d_in[16];
  const float* W_d_f   = (const float*)d_in[17];
  const float* b_d_f   = (const float*)d_in[18];
  const float* W_b_n   = (const float*)d_in[19];
  const float* b_b_n   = (const float*)d_in[20];
  const float* W_c_n   = (const float*)d_in[21];
  const float* b_c_n   = (const float*)d_in[22];
  const float* W_d_n   = (const float*)d_in[23];
  const float* b_d_n   = (const float*)d_in[24];
  const float* W_xp    = (const float*)d_in[25];
  const float* b_xp    = (const float*)d_in[26];
  const float* W_out   = (const float*)d_in[27];
  const float* b_out   = (const float*)d_in[28];
  float* out = (float*)d_out;

  // ---- workspace layout (bytes) ----
  char* ws = (char*)d_ws;
  size_t off = 0;
  auto alloc = [&](size_t bytes) { char* p = ws + off; off = (off + bytes + 255) & ~(size_t)255; return p; };
  float* G     = (float*)alloc((size_t)B_ * K_ * K_ * 4);   // 64 MB
  float* T     = (float*)alloc((size_t)ROWS_ * D_ * 4);     // 32 MB gathered tokens
  float* A1    = (float*)alloc((size_t)ROWS_ * D_ * 4);     // 32 MB (ln out / zc)
  float* A2    = (float*)alloc((size_t)ROWS_ * D_ * 4);     // 32 MB (z / head stage)
  float* y16F  = (float*)alloc((size_t)ROWS_ * S_ * 4);     // 1 MB
  float* y16N  = (float*)alloc((size_t)ROWS_ * S_ * 4);     // 1 MB
  float* yord  = (float*)alloc((size_t)ROWS_ * S_ * 4);     // 1 MB
  float* nrm   = (float*)alloc((size_t)ROWS_ * 4);
  int*   ord   = (int*)  alloc((size_t)ROWS_ * 4);
  (void)ws_size; (void)in_sizes; (void)n_in; (void)out_size;

  const dim3 blk256(256);
  const dim3 gemmGrid(D_ / 16, ROWS_ / 512);        // (32, 32)
  const dim3 gramGrid(K_ / 512, K_ / 16, B_);       // (2, 64, 16)

  // ===== Gram + norms over raw tokens (shared by FPS and NPS) =====
  gram_wmma_k<<<gramGrid, blk256, 0, stream>>>(tokens, D_, 9, G);
  row_norms_k<<<dim3(ROWS_ / 8), blk256, 0, stream>>>(tokens, D_, nrm);

  // ===== Branch F (farthest-point ordering) =====
  fps_order_k<<<dim3(B_), dim3(1024), 0, stream>>>(G, nrm, ord);
  gather_rows_k<<<dim3(ROWS_), dim3(128), 0, stream>>>(tokens, ord, T, D_);
  layernorm_k<<<dim3(ROWS_ / 8), blk256, 0, stream>>>(T, ln_f_g, ln_f_b, A1);
  gemm_wmma_k<<<gemmGrid, blk256, 0, stream>>>(A1, D_, W_lin_f, D_, b_lin_f, A2, D_, D_, 1); // z = silu
  gemm_wmma_k<<<gemmGrid, blk256, 0, stream>>>(A2, D_, W_conv_f, D_, b_conv_f, A1, D_, D_, 0); // zc
  head_gemm_k<<<dim3(ROWS_ / 128), blk256, 0, stream>>>(A1, W_b_f, b_b_f, W_c_f, b_c_f,
                                                        W_d_f, b_d_f, W_xp, b_xp, A2);
  head_pointwise_k<<<dim3((ROWS_ * S_) / 256), blk256, 0, stream>>>(A2, A1, y16F);

  // ===== Branch N (nearest-point ordering) =====
  nps_order_k<<<dim3(B_), dim3(1024), 0, stream>>>(G, nrm, ord);
  gather_rows_k<<<dim3(ROWS_), dim3(128), 0, stream>>>(tokens, ord, T, D_);
  layernorm_k<<<dim3(ROWS_ / 8), blk256, 0, stream>>>(T, ln_n_g, ln_n_b, A1);
  gemm_wmma_k<<<gemmGrid, blk256, 0, stream>>>(A1, D_, W_lin_n, D_, b_lin_n, A2, D_, D_, 1);
  gemm_wmma_k<<<gemmGrid, blk256, 0, stream>>>(A2, D_, W_conv_n, D_, b_conv_n, A1, D_, D_, 0);
  head_gemm_k<<<dim3(ROWS_ / 128), blk256, 0, stream>>>(A1, W_b_n, b_b_n, W_c_n, b_c_n,
                                                        W_d_n, b_d_n, W_xp, b_xp, A2);
  head_pointwise_k<<<dim3((ROWS_ * S_) / 256), blk256, 0, stream>>>(A2, A1, y16N);

  // ===== Mix, final FPS over 16 effective channels, output projection =====
  mix_k<<<dim3((ROWS_ * S_) / 256), blk256, 0, stream>>>(y16F, y16N);
  gram_wmma_k<<<gramGrid, blk256, 0, stream>>>(y16F, S_, 4, G);
  row_norms_k<<<dim3(ROWS_ / 8), blk256, 0, stream>>>(y16F, S_, nrm);
  fps_order_k<<<dim3(B_), dim3(1024), 0, stream>>>(G, nrm, ord);
  gather_rows_k<<<dim3(ROWS_), dim3(128), 0, stream>>>(y16F, ord, yord, S_);
  // out = y_ord16 @ W_out[0:16,:] + b_out   (rows 16..511 of y_ord are zero)
  gemm_wmma_k<<<gemmGrid, blk256, 0, stream>>>(yord, S_, W_out, D_, b_out, out, D_, S_, 0);
}